// VNNGP_74947179316106
// MI455X (gfx1250) — compile-verified
//
#include <hip/hip_runtime.h>
#include <math.h>

typedef float v2f __attribute__((ext_vector_type(2)));
typedef float v8f __attribute__((ext_vector_type(8)));

#define M_IND   512
#define DIMS    8
#define KNN     16
#define JITTER  1e-4f
#define FLT_BIG 3.4e38f

// ---------------------------------------------------------------------------
// Kernel 1: elementwise prep.
//   Lu = tril(Lu_raw,-1) + diag(exp(diag(Lu_raw)))
//   A  = var*exp(-0.5*||z_i - z_j||^2 / ls^2) + jitter*I      (= Kzz + jitter I)
// ---------------------------------------------------------------------------
__global__ void vnngp_prep(const float* __restrict__ Z,
                           const float* __restrict__ Lu_raw,
                           const float* __restrict__ log_ls,
                           const float* __restrict__ log_var,
                           float* __restrict__ Lu,
                           float* __restrict__ A) {
    int id = blockIdx.x * blockDim.x + threadIdx.x;
    if (id >= M_IND * M_IND) return;
    int i = id >> 9, j = id & 511;

    float ls  = expf(log_ls[0]);
    float var = expf(log_var[0]);
    float inv2 = 0.5f / (ls * ls);

    float v  = Lu_raw[id];
    float lu = (j < i) ? v : ((j == i) ? expf(v) : 0.0f);
    Lu[id] = lu;

    float sq = 0.f;
#pragma unroll
    for (int c = 0; c < DIMS; ++c) {
        float d = Z[i * DIMS + c] - Z[j * DIMS + c];
        sq = fmaf(d, d, sq);
    }
    float a = var * expf(-sq * inv2);
    if (i == j) a += JITTER;
    A[id] = a;
}

// ---------------------------------------------------------------------------
// Kernel 2: S = Lu * Lu^T  (512x512, K=512) via V_WMMA_F32_16X16X4_F32.
// One wave per 16x16 output tile; 128 WMMA k-steps per tile.
// A-frag: lane l -> Lu[tr + l%16][k0 + 2*(l/16) + {0,1}]   (float2)
// B-frag: lane l -> Lu[tc + l%16][k0 + 2*(l/16) + {0,1}]   (float2, = Lu^T col)
// C/D:    vgpr r -> S[M = r + 8*(l>=16)][N = l%16]
// ---------------------------------------------------------------------------
__global__ void __launch_bounds__(256)
vnngp_gram_wmma(const float* __restrict__ Lu, float* __restrict__ S) {
    int wave = (blockIdx.x * blockDim.x + threadIdx.x) >> 5;  // 0..1023
    int lane = threadIdx.x & 31;
    int tr = (wave >> 5) << 4;   // tile row base
    int tc = (wave & 31) << 4;   // tile col base
    int l16 = lane & 15;
    int kh  = lane >> 4;

    const float* arow = Lu + (tr + l16) * M_IND + 2 * kh;
    const float* brow = Lu + (tc + l16) * M_IND + 2 * kh;

    v8f c = {0.f, 0.f, 0.f, 0.f, 0.f, 0.f, 0.f, 0.f};
    for (int k0 = 0; k0 < M_IND; k0 += 4) {
        v2f a = *(const v2f*)(arow + k0);
        v2f b = *(const v2f*)(brow + k0);
        c = __builtin_amdgcn_wmma_f32_16x16x4_f32(
                /*neg_a=*/false, a, /*neg_b=*/false, b,
                /*c_mod=*/(short)0, c, /*reuse_a=*/false, /*reuse_b=*/false);
    }

    int mrow = tr + (kh ? 8 : 0);
#pragma unroll
    for (int r = 0; r < 8; ++r)
        S[(mrow + r) * M_IND + tc + l16] = c[r];
}

// ---------------------------------------------------------------------------
// Kernel 3: fused per-point pipeline. 16 lanes cooperate on one query point
// (2 points per wave32, 16 points per 256-thread block).
//   - 512 squared distances (Z staged in LDS, stride-9 pad -> no bank conflicts)
//   - top-16 selection (iterated group arg-min via shfl_xor within 16 lanes)
//   - lane i owns row i of augmented [B | kxz]; Gaussian elimination +
//     back-substitution via lane broadcasts -> w = B^{-1} kxz
//   - mean = w.mu_g ;  cov = var - (w.kxz - jitter*|w|^2) + w^T S_g w
// ---------------------------------------------------------------------------
__global__ void __launch_bounds__(256)
vnngp_main(const float* __restrict__ X,
           const float* __restrict__ Z,
           const float* __restrict__ mu,
           const float* __restrict__ log_ls,
           const float* __restrict__ log_var,
           const float* __restrict__ A,
           const float* __restrict__ S,
           float* __restrict__ out, int N) {
    __shared__ float zs[M_IND * 9];   // padded stride 9 (gcd(9,64)=1)
    __shared__ float mus[M_IND];

    int tid = threadIdx.x;
    for (int i = tid; i < M_IND * DIMS; i += 256) {
        int r = i >> 3, c = i & 7;
        zs[r * 9 + c] = Z[i];
    }
    for (int i = tid; i < M_IND; i += 256) mus[i] = mu[i];
    __syncthreads();

    float ls  = expf(log_ls[0]);
    float var = expf(log_var[0]);
    float inv2 = 0.5f / (ls * ls);

    int grp    = tid >> 4;         // 0..15 : point within block
    int lane16 = tid & 15;         // 0..15 : row within point
    int gbase  = tid & 16;         // base lane of 16-lane group inside wave
    int n = blockIdx.x * 16 + grp;

    // query point
    const float4* xp = (const float4*)(X + (size_t)n * DIMS);
    float4 x0 = xp[0], x1 = xp[1];
    float x[8] = {x0.x, x0.y, x0.z, x0.w, x1.x, x1.y, x1.z, x1.w};

    // --- distances: lane l handles z indices l, l+16, ..., l+496 ---
    float dl[32];
#pragma unroll
    for (int t = 0; t < 32; ++t) {
        const float* zr = &zs[(lane16 + (t << 4)) * 9];
        float sq = 0.f;
#pragma unroll
        for (int c = 0; c < DIMS; ++c) {
            float d = x[c] - zr[c];
            sq = fmaf(d, d, sq);
        }
        dl[t] = sq;
    }

    // --- top-16 selection: lane r ends up owning the r-th nearest ---
    float myd = 0.f; int myidx = 0;
#pragma unroll
    for (int r = 0; r < KNN; ++r) {
        float bm = FLT_BIG; int bt = 0;
#pragma unroll
        for (int t = 0; t < 32; ++t)
            if (dl[t] < bm) { bm = dl[t]; bt = t; }
        int bz = lane16 + (bt << 4);
#pragma unroll
        for (int m = 8; m >= 1; m >>= 1) {
            float od = __shfl_xor(bm, m, 32);
            int   oz = __shfl_xor(bz, m, 32);
            if (od < bm || (od == bm && oz < bz)) { bm = od; bz = oz; }
        }
        if (lane16 == r) { myd = bm; myidx = bz; }
        int ow = bz & 15, slot = bz >> 4;
#pragma unroll
        for (int t = 0; t < 32; ++t)
            if (lane16 == ow && t == slot) dl[t] = FLT_BIG;
    }

    // broadcast the 16 selected indices to all lanes of the group
    int idxs[16];
#pragma unroll
    for (int j = 0; j < 16; ++j) idxs[j] = __shfl(myidx, gbase + j, 32);

    float kxz = var * expf(-myd * inv2);

    const float* Arow = A + (size_t)myidx * M_IND;
    const float* Srow = S + (size_t)myidx * M_IND;
    __builtin_prefetch(Srow, 0, 1);   // global_prefetch_b8

    // augmented row [B | kxz]; B = A_gathered + jitter on diag (=> Kzz_g + 2*jitter I)
    float row[17];
#pragma unroll
    for (int j = 0; j < 16; ++j)
        row[j] = Arow[idxs[j]] + ((j == lane16) ? JITTER : 0.f);
    row[16] = kxz;

    // --- forward elimination (SPD, no pivoting) ---
#pragma unroll
    for (int k = 0; k < 16; ++k) {
        float pk = __shfl(row[k], gbase + k, 32);
        float f  = row[k] / pk;
        bool act = (lane16 > k);
#pragma unroll
        for (int j = k + 1; j < 17; ++j) {
            float pj = __shfl(row[j], gbase + k, 32);
            if (act) row[j] = fmaf(-f, pj, row[j]);
        }
    }
    // --- back substitution: lane i ends with w_i ---
    float w = 0.f;
#pragma unroll
    for (int k = 15; k >= 0; --k) {
        float xk = __shfl(row[16] / row[k], gbase + k, 32);
        if (lane16 == k) w = xk;
        if (lane16 < k) row[16] = fmaf(-row[k], xk, row[16]);
    }

    // --- outputs ---
    float q2part = 0.f;
#pragma unroll
    for (int j = 0; j < 16; ++j) {
        float wj = __shfl(w, gbase + j, 32);
        q2part = fmaf(Srow[idxs[j]], wj, q2part);
    }
    q2part *= w;                       // contributes w_i * (S_g w)_i
    float mpart  = w * mus[myidx];     // mean contribution
    float q1part = w * kxz;            // w^T B w  (since B w = kxz)
    float wnpart = w * w;
#pragma unroll
    for (int m = 8; m >= 1; m >>= 1) {
        mpart  += __shfl_xor(mpart,  m, 32);
        q1part += __shfl_xor(q1part, m, 32);
        wnpart += __shfl_xor(wnpart, m, 32);
        q2part += __shfl_xor(q2part, m, 32);
    }
    if (lane16 == 0) {
        // w^T little_Kzz w = w^T (B - jitter I) w = q1 - jitter*|w|^2
        float cov = var - (q1part - JITTER * wnpart) + q2part;
        float sd  = sqrtf(fmaxf(cov, 0.05f));
        out[n]     = mpart;
        out[N + n] = sd;
    }
}

// ---------------------------------------------------------------------------
// Host launcher (graph-capture safe: only kernel launches on `stream`).
// Workspace layout: Lu (1MB) | A (1MB) | S (1MB).
// ---------------------------------------------------------------------------
extern "C" void kernel_launch(void* const* d_in, const int* in_sizes, int n_in,
                              void* d_out, int out_size, void* d_ws, size_t ws_size,
                              hipStream_t stream) {
    const float* X      = (const float*)d_in[0];
    const float* Z      = (const float*)d_in[1];
    const float* Lu_raw = (const float*)d_in[2];
    const float* mu     = (const float*)d_in[3];
    const float* logls  = (const float*)d_in[4];
    const float* logvar = (const float*)d_in[5];
    int N = in_sizes[0] / DIMS;

    float* ws = (float*)d_ws;
    float* Lu = ws;
    float* A  = ws + M_IND * M_IND;
    float* S  = ws + 2 * M_IND * M_IND;

    vnngp_prep<<<(M_IND * M_IND + 255) / 256, 256, 0, stream>>>(
        Z, Lu_raw, logls, logvar, Lu, A);

    // 32x32 = 1024 tiles, 8 waves (256 thr) per block -> 128 blocks
    vnngp_gram_wmma<<<128, 256, 0, stream>>>(Lu, S);

    vnngp_main<<<N / 16, 256, 0, stream>>>(
        X, Z, mu, logls, logvar, A, S, (float*)d_out, N);
}